// SmagorinskyModel_44341242363984
// MI455X (gfx1250) — compile-verified
//
#include <hip/hip_runtime.h>
#include <math.h>
#include <stdint.h>

// Smagorinsky SGS stress, velocity [2,3,128,128,128] f32 -> tau [2,6,128,128,128] f32.
// Pure bandwidth problem (~151 MB compulsory traffic, ~6.5 us at 23.3 TB/s).
// CDNA5 path used: async global->LDS copies (ASYNCcnt) for the x-row staging,
// wide b128 loads for y/z neighbors (L2-resident reuse), nontemporal b128 stores.

typedef float v4f __attribute__((ext_vector_type(4)));

#define NN 128
#define PLANE (NN * NN)       // 16384
#define VOL (NN * NN * NN)    // 2097152
#define ROWS_PER_BLOCK 8

__device__ __forceinline__ void async_copy_b128(uint32_t lds_byte_off, const float* gptr) {
  // CDNA5: global -> LDS copy at byte offset vdst, 16 bytes per lane, tracked by ASYNCcnt.
  asm volatile("global_load_async_to_lds_b128 %0, %1, off"
               :
               : "v"(lds_byte_off), "v"(gptr)
               : "memory");
}

__device__ __forceinline__ void wait_async_lds() {
  asm volatile("s_wait_asynccnt 0" ::: "memory");
}

__global__ __launch_bounds__(256) void smag_kernel(const float* __restrict__ vel,
                                                   float* __restrict__ tau) {
  __shared__ float sm[3][ROWS_PER_BLOCK][NN];   // 12 KB: center x-rows of u,v,w

  const int t  = threadIdx.x;
  const int r  = t >> 5;        // row in tile (0..7) == wave id
  const int l  = t & 31;        // lane
  const int x0 = l << 2;        // x start (multiple of 4)

  int blk = blockIdx.x;
  const int ytile = blk & 15;  blk >>= 4;
  const int z     = blk & 127; blk >>= 7;
  const int b     = blk;                   // 0..1
  const int y     = ytile * ROWS_PER_BLOCK + r;

  const int zm1 = (z + NN - 1) & (NN - 1);
  const int zp1 = (z + 1) & (NN - 1);
  const int ym1 = (y + NN - 1) & (NN - 1);
  const int yp1 = (y + 1) & (NN - 1);

  // ---- Stage center rows (needed for the +-1 x-shift) into LDS asynchronously.
  #pragma unroll
  for (int c = 0; c < 3; ++c) {
    const float* g = vel + (size_t)(b * 3 + c) * VOL + z * PLANE + y * NN + x0;
    const uint32_t lo = (uint32_t)(uintptr_t)&sm[c][r][x0];  // low 32 bits = LDS offset
    async_copy_b128(lo, g);
  }

  // ---- Issue y/z neighbor loads (overlap with the async LDS copies).
  v4f ymv[3], ypv[3], zmv[3], zpv[3];
  #pragma unroll
  for (int c = 0; c < 3; ++c) {
    const float* cb = vel + (size_t)(b * 3 + c) * VOL;
    ymv[c] = *(const v4f*)(cb + z   * PLANE + ym1 * NN + x0);
    ypv[c] = *(const v4f*)(cb + z   * PLANE + yp1 * NN + x0);
    zmv[c] = *(const v4f*)(cb + zm1 * PLANE + y   * NN + x0);
    zpv[c] = *(const v4f*)(cb + zp1 * PLANE + y   * NN + x0);
  }

  wait_async_lds();
  __syncthreads();

  const float inv2dx = 10.185916357881302f;   // N / (4*pi) = 1/(2*dx), dx = 2*pi/N

  // du[i][j]: j = 0 -> axis2 (z), 1 -> axis3 (y), 2 -> axis4 (x) to match reference.
  v4f ddx[3], ddy[3], ddz[3];
  #pragma unroll
  for (int c = 0; c < 3; ++c) {
    const v4f   ctr   = *(const v4f*)&sm[c][r][x0];
    const float left  = sm[c][r][(x0 + NN - 1) & (NN - 1)];
    const float right = sm[c][r][(x0 + 4) & (NN - 1)];
    v4f dxv;
    dxv.x = ctr.y - left;
    dxv.y = ctr.z - ctr.x;
    dxv.z = ctr.w - ctr.y;
    dxv.w = right - ctr.z;
    ddx[c] = dxv * inv2dx;
    ddy[c] = (ypv[c] - ymv[c]) * inv2dx;
    ddz[c] = (zpv[c] - zmv[c]) * inv2dx;
  }

  // Channels: [S00, S11, S22, S01, S02, S12] with du[i][0]=d/dz, du[i][1]=d/dy, du[i][2]=d/dx
  const v4f S0 = ddz[0];
  const v4f S1 = ddy[1];
  const v4f S2 = ddx[2];
  const v4f S3 = 0.5f * (ddy[0] + ddz[1]);
  const v4f S4 = 0.5f * (ddx[0] + ddz[2]);
  const v4f S5 = 0.5f * (ddx[1] + ddy[2]);

  const v4f ss = S0 * S0 + S1 * S1 + S2 * S2 + S3 * S3 + S4 * S4 + S5 * S5;
  v4f mag;
  #pragma unroll
  for (int k = 0; k < 4; ++k) mag[k] = __builtin_sqrtf(2.0f * ss[k]);

  const float coef = -0.0648f;                // -2 * CS^2 * FILTER_WIDTH^2, CS = 0.18
  const v4f kf = coef * mag;

  float* ob = tau + (size_t)(b * 6) * VOL + z * PLANE + y * NN + x0;
  __builtin_nontemporal_store(kf * S0, (v4f*)(ob + 0 * VOL));
  __builtin_nontemporal_store(kf * S1, (v4f*)(ob + 1 * VOL));
  __builtin_nontemporal_store(kf * S2, (v4f*)(ob + 2 * VOL));
  __builtin_nontemporal_store(kf * S3, (v4f*)(ob + 3 * VOL));
  __builtin_nontemporal_store(kf * S4, (v4f*)(ob + 4 * VOL));
  __builtin_nontemporal_store(kf * S5, (v4f*)(ob + 5 * VOL));
}

extern "C" void kernel_launch(void* const* d_in, const int* in_sizes, int n_in,
                              void* d_out, int out_size, void* d_ws, size_t ws_size,
                              hipStream_t stream) {
  (void)in_sizes; (void)n_in; (void)d_ws; (void)ws_size; (void)out_size;
  const float* vel = (const float*)d_in[0];
  float* tau = (float*)d_out;
  const int blocks = 2 * NN * (NN / ROWS_PER_BLOCK);  // 2 * 128 * 16 = 4096
  smag_kernel<<<blocks, 256, 0, stream>>>(vel, tau);
}